// BatchedLUTNodes_29678224015513
// MI455X (gfx1250) — compile-verified
//
#include <hip/hip_runtime.h>
#include <stdint.h>

// ---------------------------------------------------------------------------
// BatchedLUTNodes: out[b,n] = sum_e prod_i (bit_i(e)?x_i:1-x_i) * tables[n,e]
// == multilinear interpolation over a 6-cube -> binary fold, 63 lerps/output.
//
// Memory-bound on MI455X (~30 MB @ 23.3 TB/s ~ 1.3 us); fold keeps VALU at
// ~110 flops/output so compute stays under the bandwidth roof. Tables are
// staged per 32-node tile into LDS via the CDNA5 async global->LDS path
// (ASYNCcnt), read back conflict-free as pair-major float2 (ds_load_b64).
// ---------------------------------------------------------------------------

#define AS1 __attribute__((address_space(1)))
#define AS3 __attribute__((address_space(3)))

typedef int v2i __attribute__((ext_vector_type(2)));

constexpr int kBatch   = 128;
constexpr int kNodes   = 8192;
constexpr int kEntries = 64;   // 2^6
constexpr int kNTile   = 32;   // nodes per block (== lanes per wave)
constexpr int kThreads = 256;  // 8 waves (wave32)

__device__ __forceinline__ float clamp01(float v) {
  return fminf(fmaxf(v, 0.0f), 1.0f);
}

__device__ __forceinline__ float2 lerp2(float2 a, float2 b, float2 t) {
  float2 r;
  r.x = fmaf(t.x, b.x - a.x, a.x);
  r.y = fmaf(t.y, b.y - a.y, a.y);
  return r;
}

__global__ __launch_bounds__(kThreads)
void lut_fold_kernel(const float* __restrict__ x,
                     const float* __restrict__ tables,
                     float* __restrict__ out) {
  // LDS tile: pair-major. smem[p * 32 + n] = (tables[n0+n][2p], tables[n0+n][2p+1])
  // ds_load_b64 at dword (64p + 2*lane): each lane owns a distinct bank pair.
  __shared__ float2 smem[kNTile * (kEntries / 2)];  // 8 KB

  const int tid  = (int)threadIdx.x;
  const int nl   = tid & 31;         // node-in-tile == lane id
  const int w    = tid >> 5;         // wave id 0..7 -> batch offset
  const int n0   = (int)blockIdx.x * kNTile;
  const int b0   = (int)blockIdx.y * 32;   // 32 batches per block
  const int node = n0 + nl;

  // ---- stage the 32x64 float table tile into LDS (async global -> LDS) ----
  // pair index within tile: g = n*32 + p  (global side is contiguous in g)
  const float2* tab2 = reinterpret_cast<const float2*>(tables) + (size_t)n0 * (kEntries / 2);
#pragma unroll
  for (int k = 0; k < 4; ++k) {
    const int g = tid + k * kThreads;       // 0..1023
    const int n = g >> 5;
    const int p = g & 31;
    float2* src = const_cast<float2*>(tab2 + g);
    float2* dst = &smem[p * kNTile + n];
#if __has_builtin(__builtin_amdgcn_global_load_async_to_lds_b64)
    __builtin_amdgcn_global_load_async_to_lds_b64(
        (AS1 v2i*)src, (AS3 v2i*)dst, /*offset=*/0, /*cpol=*/0);
#else
    *dst = *src;                            // sync fallback
#endif
  }
#if __has_builtin(__builtin_amdgcn_s_wait_asynccnt)
  __builtin_amdgcn_s_wait_asynccnt(0);
#else
  asm volatile("s_wait_asynccnt 0" ::: "memory");
#endif
  __syncthreads();

  // ---- compute: each thread folds two batches packed in float2 ----
#pragma unroll
  for (int it = 0; it < 2; ++it) {
    const int bA = b0 + it * 16 + w;   // batches b0..b0+31 across 8 waves x 2 iters
    const int bB = bA + 8;

    // x records are 24 B (6 floats), 8B-aligned -> three b64 loads each,
    // contiguous across lanes (lane == consecutive node).
    const float2* xA = reinterpret_cast<const float2*>(x + ((size_t)bA * kNodes + node) * 6);
    const float2* xB = reinterpret_cast<const float2*>(x + ((size_t)bB * kNodes + node) * 6);
    const float2 ua0 = xA[0], ua1 = xA[1], ua2 = xA[2];
    const float2 ub0 = xB[0], ub1 = xB[1], ub2 = xB[2];

    float2 Xs[6];
    Xs[0] = make_float2(clamp01(ua0.x), clamp01(ub0.x));
    Xs[1] = make_float2(clamp01(ua0.y), clamp01(ub0.y));
    Xs[2] = make_float2(clamp01(ua1.x), clamp01(ub1.x));
    Xs[3] = make_float2(clamp01(ua1.y), clamp01(ub1.y));
    Xs[4] = make_float2(clamp01(ua2.x), clamp01(ub2.x));
    Xs[5] = make_float2(clamp01(ua2.y), clamp01(ub2.y));

    // Level 0: fold bit 0 directly out of LDS. The (t.y - t.x) sub is shared
    // between the two packed batches.
    float2 acc[32];
#pragma unroll
    for (int p = 0; p < 32; ++p) {
      const float2 t = smem[p * kNTile + nl];   // ds_load_b64, conflict-free
      const float d = t.y - t.x;
      acc[p].x = fmaf(Xs[0].x, d, t.x);
      acc[p].y = fmaf(Xs[0].y, d, t.x);
    }

    // Levels 1..5: fold bits 1..5 (16 -> 8 -> 4 -> 2 -> 1 survivors).
#pragma unroll
    for (int lvl = 1; lvl < 6; ++lvl) {
      const int len = 32 >> lvl;
#pragma unroll
      for (int j = 0; j < 16; ++j) {
        if (j < len) acc[j] = lerp2(acc[2 * j], acc[2 * j + 1], Xs[lvl]);
      }
    }

    out[(size_t)bA * kNodes + node] = acc[0].x;   // coalesced b32 stores
    out[(size_t)bB * kNodes + node] = acc[0].y;
  }
}

extern "C" void kernel_launch(void* const* d_in, const int* in_sizes, int n_in,
                              void* d_out, int out_size, void* d_ws, size_t ws_size,
                              hipStream_t stream) {
  (void)in_sizes; (void)n_in; (void)d_ws; (void)ws_size; (void)out_size;
  const float* x      = (const float*)d_in[0];   // (128, 8192, 6) f32
  const float* tables = (const float*)d_in[1];   // (8192, 64)    f32
  float* out          = (float*)d_out;           // (128, 8192)   f32

  dim3 grid(kNodes / kNTile, kBatch / 32);       // (256, 4)
  lut_fold_kernel<<<grid, dim3(kThreads), 0, stream>>>(x, tables, out);
}